// GraphConvBlock_v2_47321949667568
// MI455X (gfx1250) — compile-verified
//
#include <hip/hip_runtime.h>
#include <math.h>

typedef __attribute__((ext_vector_type(2))) float v2f;
typedef __attribute__((ext_vector_type(8))) float v8f;

#define DIM 512
#define NT 4  // N-tiles (of 16) per wave: 16x64 output slab, A-frag reused 4x

// ---------------------------------------------------------------------------
// GEMM: XW[i][j] = sum_k X[i][k] * W[j][k]   (i.e. X @ W^T), f32 WMMA 16x16x4
// One wave -> 16 rows x 64 cols of XW. Grid: (Nrows/16) * (DIM/64) waves.
// ---------------------------------------------------------------------------
__global__ __launch_bounds__(256) void gcn_gemm_xwT(const float* __restrict__ X,
                                                    const float* __restrict__ W,
                                                    float* __restrict__ XW,
                                                    int Nrows) {
  const int wave = (blockIdx.x * blockDim.x + threadIdx.x) >> 5;
  const int lane = threadIdx.x & 31;
  const int tilesN = DIM / (16 * NT);        // 8
  const int tm = wave / tilesN;              // M tile (16 rows)
  const int tn = (wave % tilesN) * NT;       // first N tile (of 16 cols)
  if (tm * 16 >= Nrows) return;              // wave-uniform (EXEC stays all-1s)

  const int half = lane >> 4;                // 0: K={0,1}, 1: K={2,3}
  const int l16  = lane & 15;

  const float* xrow = X + (size_t)(tm * 16 + l16) * DIM + half * 2;
  const float* wrow = W + (size_t)(tn * 16 + l16) * DIM + half * 2;

  v8f c[NT];
#pragma unroll
  for (int t = 0; t < NT; ++t) c[t] = (v8f){0.f,0.f,0.f,0.f,0.f,0.f,0.f,0.f};

  for (int k0 = 0; k0 < DIM; k0 += 4) {
    const v2f a = *(const v2f*)(xrow + k0);
#pragma unroll
    for (int t = 0; t < NT; ++t) {
      const v2f b = *(const v2f*)(wrow + (size_t)t * 16 * DIM + k0);
      c[t] = __builtin_amdgcn_wmma_f32_16x16x4_f32(
          /*neg_a=*/false, a, /*neg_b=*/false, b,
          /*c_mod=*/(short)0, c[t], /*reuse_a=*/false, /*reuse_b=*/false);
    }
  }

  // C/D layout: VGPR r, lane l -> M = r + 8*(l>=16), N = l%16
#pragma unroll
  for (int t = 0; t < NT; ++t) {
    float* orow = XW + (size_t)(tm * 16 + half * 8) * DIM + (tn + t) * 16 + l16;
#pragma unroll
    for (int r = 0; r < 8; ++r) orow[(size_t)r * DIM] = c[t][r];
  }
}

// ---------------------------------------------------------------------------
// init: deg[i] = 1.0 (self loop), stats (sums/sumsq/scale/shift) = 0
// ---------------------------------------------------------------------------
__global__ void gcn_init(float* __restrict__ deg, float* __restrict__ stats, int n) {
  int i = blockIdx.x * blockDim.x + threadIdx.x;
  if (i < n) deg[i] = 1.0f;
  if (i < 2048) stats[i] = 0.0f;
}

__global__ void gcn_deg_count(const int* __restrict__ col, float* __restrict__ deg, int E) {
  int e = blockIdx.x * blockDim.x + threadIdx.x;
  if (e < E) atomicAdd(&deg[col[e]], 1.0f);
}

__global__ void gcn_deg_rsqrt(float* __restrict__ deg, int n) {
  int i = blockIdx.x * blockDim.x + threadIdx.x;
  if (i < n) deg[i] = rsqrtf(deg[i]);   // deg >= 1 always (self loop)
}

// ---------------------------------------------------------------------------
// h[i][:] = dinv[i]^2 * xw[i][:]   (self-loop contribution; also inits h)
// ---------------------------------------------------------------------------
__global__ void gcn_selfloop_init(const float* __restrict__ xw,
                                  const float* __restrict__ dinv,
                                  float* __restrict__ h, int total4) {
  int idx = blockIdx.x * blockDim.x + threadIdx.x;  // over N*DIM/4
  if (idx >= total4) return;
  int i = idx >> 7;                                  // 128 float4 per row
  float s = dinv[i]; s *= s;
  float4 v = ((const float4*)xw)[idx];
  v.x *= s; v.y *= s; v.z *= s; v.w *= s;
  ((float4*)h)[idx] = v;
}

// ---------------------------------------------------------------------------
// Edge scatter: h[col[e]][:] += dinv[row]*dinv[col] * xw[row[e]][:]
// 128 threads per edge, float4 per thread; atomics land mostly in 192MB L2.
// ---------------------------------------------------------------------------
__global__ __launch_bounds__(256) void gcn_scatter(const int* __restrict__ rows,
                                                   const int* __restrict__ cols,
                                                   const float* __restrict__ dinv,
                                                   const float* __restrict__ xw,
                                                   float* __restrict__ h, int E) {
  unsigned tid = blockIdx.x * blockDim.x + threadIdx.x;
  int e = tid >> 7;
  if (e >= E) return;
  int c4 = (tid & 127) << 2;
  int r = rows[e], c = cols[e];
  float nrm = dinv[r] * dinv[c];
  float4 v = *(const float4*)(xw + (size_t)r * DIM + c4);
  float* dst = h + (size_t)c * DIM + c4;
  atomicAdd(dst + 0, nrm * v.x);
  atomicAdd(dst + 1, nrm * v.y);
  atomicAdd(dst + 2, nrm * v.z);
  atomicAdd(dst + 3, nrm * v.w);
}

// ---------------------------------------------------------------------------
// BN column stats: each block accumulates 256 rows in registers (coalesced),
// then 2 atomics per column.
// ---------------------------------------------------------------------------
__global__ __launch_bounds__(256) void gcn_bn_stats(const float* __restrict__ h,
                                                    float* __restrict__ sums,
                                                    float* __restrict__ sumsq,
                                                    int Nrows) {
  int r0 = blockIdx.x * 256;
  int t = threadIdx.x;
  int rend = min(r0 + 256, Nrows);
  float s0 = 0.f, s1 = 0.f, q0 = 0.f, q1 = 0.f;
  for (int i = r0; i < rend; ++i) {
    float a = h[(size_t)i * DIM + t];
    float b = h[(size_t)i * DIM + 256 + t];
    s0 += a; q0 += a * a;
    s1 += b; q1 += b * b;
  }
  atomicAdd(&sums[t],        s0);
  atomicAdd(&sums[t + 256],  s1);
  atomicAdd(&sumsq[t],       q0);
  atomicAdd(&sumsq[t + 256], q1);
}

__global__ void gcn_bn_finalize(const float* __restrict__ sums,
                                const float* __restrict__ sumsq,
                                const float* __restrict__ gamma,
                                const float* __restrict__ beta,
                                float* __restrict__ scale,
                                float* __restrict__ shift, int Nrows) {
  int j = threadIdx.x;  // 512 threads, 1 block
  float invn = 1.0f / (float)Nrows;
  float mean = sums[j] * invn;
  float var  = sumsq[j] * invn - mean * mean;   // biased, matches torch BN
  float sc = gamma[j] * rsqrtf(var + 1e-5f);
  scale[j] = sc;
  shift[j] = beta[j] - mean * sc;
}

// ---------------------------------------------------------------------------
// out = gelu_exact(h*scale + shift) + x   (in place on h == d_out)
// ---------------------------------------------------------------------------
__global__ void gcn_bn_gelu_res(float* __restrict__ h,
                                const float* __restrict__ x,
                                const float* __restrict__ scale,
                                const float* __restrict__ shift, int total4) {
  int idx = blockIdx.x * blockDim.x + threadIdx.x;  // over N*DIM/4
  if (idx >= total4) return;
  int j = (idx & 127) << 2;
  float4 v = ((float4*)h)[idx];
  float4 xr = ((const float4*)x)[idx];
  const float k = 0.70710678118654752440f;  // 1/sqrt(2)
  float a0 = v.x * scale[j + 0] + shift[j + 0];
  float a1 = v.y * scale[j + 1] + shift[j + 1];
  float a2 = v.z * scale[j + 2] + shift[j + 2];
  float a3 = v.w * scale[j + 3] + shift[j + 3];
  v.x = 0.5f * a0 * (1.0f + erff(a0 * k)) + xr.x;
  v.y = 0.5f * a1 * (1.0f + erff(a1 * k)) + xr.y;
  v.z = 0.5f * a2 * (1.0f + erff(a2 * k)) + xr.z;
  v.w = 0.5f * a3 * (1.0f + erff(a3 * k)) + xr.w;
  ((float4*)h)[idx] = v;
}

// ---------------------------------------------------------------------------
extern "C" void kernel_launch(void* const* d_in, const int* in_sizes, int n_in,
                              void* d_out, int out_size, void* d_ws, size_t ws_size,
                              hipStream_t stream) {
  const float* x     = (const float*)d_in[0];
  const int*   ei    = (const int*)d_in[1];     // [2,E] flat: rows then cols
  const float* W     = (const float*)d_in[2];
  // d_in[3] = b (cancels exactly under BN mean subtraction -> skipped)
  const float* gamma = (const float*)d_in[4];
  const float* beta  = (const float*)d_in[5];

  const int N = in_sizes[0] / DIM;        // 50000
  const int E = in_sizes[1] / 2;          // 800000
  const int* rows = ei;
  const int* cols = ei + E;

  // workspace layout
  float* xw    = (float*)d_ws;            // N*DIM
  float* deg   = xw + (size_t)N * DIM;    // N   (becomes dinv in place)
  float* stats = deg + N;                 // 2048: sums | sumsq | scale | shift
  float* sums  = stats;
  float* sumsq = stats + 512;
  float* scale = stats + 1024;
  float* shift = stats + 1536;

  float* h = (float*)d_out;               // accumulator lives in d_out

  const int total4 = (N * DIM) / 4;

  // 1) init deg=1 (self loops) and zero stats
  {
    int n = N > 2048 ? N : 2048;
    gcn_init<<<(n + 255) / 256, 256, 0, stream>>>(deg, stats, N);
  }
  // 2) XW = X @ W^T via f32 WMMA
  {
    int waves = (N / 16) * (DIM / (16 * NT));   // 3125 * 8
    gcn_gemm_xwT<<<(waves * 32 + 255) / 256, 256, 0, stream>>>(x, W, xw, N);
  }
  // 3) degree over targets, 4) dinv = rsqrt(deg)
  gcn_deg_count<<<(E + 255) / 256, 256, 0, stream>>>(cols, deg, E);
  gcn_deg_rsqrt<<<(N + 255) / 256, 256, 0, stream>>>(deg, N);
  // 5) h = dinv^2 * xw   (self-loop term, initializes d_out)
  gcn_selfloop_init<<<(total4 + 255) / 256, 256, 0, stream>>>(xw, deg, h, total4);
  // 6) edge scatter-add (L2-resident atomics)
  {
    long long work = (long long)E * 128;
    gcn_scatter<<<(unsigned)((work + 255) / 256), 256, 0, stream>>>(rows, cols, deg, xw, h, E);
  }
  // 7) BN stats, 8) scale/shift
  gcn_bn_stats<<<(N + 255) / 256, 256, 0, stream>>>(h, sums, sumsq, N);
  gcn_bn_finalize<<<1, 512, 0, stream>>>(sums, sumsq, gamma, beta, scale, shift, N);
  // 9) fused BN * GELU(erf) + residual, in place on d_out
  gcn_bn_gelu_res<<<(total4 + 255) / 256, 256, 0, stream>>>(h, x, scale, shift, total4);
}